// GCN_Variant_5463198401300
// MI455X (gfx1250) — compile-verified
//
#include <hip/hip_runtime.h>
#include <hip/hip_bf16.h>

// ---------------------------------------------------------------------------
// GCN (3-layer) for MI455X / gfx1250.
//
// Roofline: GEMMs total ~4.3 GFLOP (trivial); edge aggregation moves ~3.5 GB
// through L2 (all buffers fit the 192 MB L2) -> memory/atomic bound.
// => keep exact fp32 precision, use V_WMMA_F32_16X16X4_F32 for the dense
// transforms, and optimize the gather/scatter phase (b128 gathers, f32
// atomics). One wave32 owns a 16-row output strip and all column tiles;
// per k-step: one A-fragment load + a clause of TN B-fragment loads into
// distinct registers, then TN back-to-back WMMAs (no per-WMMA load waits).
// ---------------------------------------------------------------------------

typedef float v2f __attribute__((ext_vector_type(2)));
typedef float v8f __attribute__((ext_vector_type(8)));

// ---------------- degree / normalization ----------------

__global__ void deg_init_kernel(float* __restrict__ deg, int n) {
    int i = blockIdx.x * blockDim.x + threadIdx.x;
    if (i < n) deg[i] = 1.0f;   // self-loop contributes 1 to every node
}

__global__ void deg_count_kernel(const int* __restrict__ dst,
                                 float* __restrict__ deg, int ne) {
    int e = blockIdx.x * blockDim.x + threadIdx.x;
    if (e < ne) atomicAdd(&deg[dst[e]], 1.0f);
}

__global__ void deg_to_dinv_kernel(float* __restrict__ deg, int n) {
    int i = blockIdx.x * blockDim.x + threadIdx.x;
    if (i < n) {
        float d = deg[i];
        deg[i] = (d > 0.0f) ? rsqrtf(d) : 0.0f;
    }
}

// ---------------- dense transform: out[M,16*TN] = (relu?)A[M,K] @ W --------
//
// One wave32 computes one 16-row strip x TN column tiles with
// V_WMMA_F32_16X16X4_F32. ISA 7.12.2 layouts (32-bit operands, wave32):
//   A 16x4 : lane&15 = M row; lanes 0-15 hold K={k,k+1} in {v0,v1},
//            lanes 16-31 hold K={k+2,k+3}.
//   B 4x16 : lane&15 = N col; same K split across the two lane halves.
//   D 16x16: 8 VGPRs; n = lane&15, m = vgpr + 8*(lane>>4).

template <bool RELU, int TN>
__global__ void gemm_wmma_strip(const float* __restrict__ A,
                                const float* __restrict__ W,
                                float* __restrict__ out,
                                int M, int K) {
    const int Nc   = TN * 16;
    const int wave = (blockIdx.x * blockDim.x + threadIdx.x) >> 5;
    const int lane = threadIdx.x & 31;

    const int tm = wave;                 // 16-row strip index
    if (tm * 16 >= M) return;

    const int ml    = lane & 15;         // A row / B,D col within tile
    const int khalf = (lane >> 4) << 1;  // 0 or 2

    const float* arow = A + (size_t)(tm * 16 + ml) * K;

    v8f acc[TN];
#pragma unroll
    for (int t = 0; t < TN; ++t) acc[t] = (v8f){};

    for (int k = 0; k < K; k += 4) {
        // A fragment: one b64 load (8B aligned), shared by all TN tiles
        v2f a = *(const v2f*)(arow + k + khalf);
        if (RELU) {
            a.x = fmaxf(a.x, 0.0f);
            a.y = fmaxf(a.y, 0.0f);
        }

        // Preload ALL B fragments into distinct registers first, so the
        // backend emits one load clause + one wait, then a WMMA burst.
        const float* w0 = W + (size_t)(k + khalf) * Nc + ml;
        v2f b[TN];
#pragma unroll
        for (int t = 0; t < TN; ++t) {
            b[t].x = w0[t * 16];
            b[t].y = w0[Nc + t * 16];
        }

#pragma unroll
        for (int t = 0; t < TN; ++t) {
            acc[t] = __builtin_amdgcn_wmma_f32_16x16x4_f32(
                /*neg_a=*/false, a, /*neg_b=*/false, b[t],
                /*c_mod=*/(short)0, acc[t], /*reuse_a=*/false, /*reuse_b=*/false);
        }
    }

    const int row0 = tm * 16 + ((lane >> 4) << 3);  // m = vgpr + 8*(lane>=16)
#pragma unroll
    for (int t = 0; t < TN; ++t) {
        float* o = out + (size_t)row0 * Nc + t * 16 + ml;
#pragma unroll
        for (int r = 0; r < 8; ++r) o[(size_t)r * Nc] = acc[t][r];
    }
}

// ---------------- aggregation ----------------

// agg[i][c] = bias[c] + dinv[i]^2 * t[i][c]   (self-loop term + bias)
// Writes EVERY element -> also clears the 0xAA poison in the target buffer.
__global__ void self_init_kernel(const float* __restrict__ t,
                                 const float* __restrict__ dinv,
                                 const float* __restrict__ bias,
                                 float* __restrict__ agg,
                                 int n, int C) {
    int idx = blockIdx.x * blockDim.x + threadIdx.x;
    if (idx >= n * C) return;
    int i = idx / C;
    int c = idx - i * C;
    float di = dinv[i];
    agg[idx] = bias[c] + di * di * t[idx];
}

// One wave32 per edge. Gather the whole src row with b128/b64 loads
// (rows are 512B/256B aligned), scatter with f32 atomics into dst.
template <int C>
__global__ void edge_agg_kernel(const float* __restrict__ t,
                                const float* __restrict__ dinv,
                                const int* __restrict__ src,
                                const int* __restrict__ dst,
                                float* __restrict__ agg, int ne) {
    int e = (blockIdx.x * blockDim.x + threadIdx.x) >> 5;
    if (e >= ne) return;
    int lane = threadIdx.x & 31;

    int s = src[e];
    int d = dst[e];
    float nrm = dinv[s] * dinv[d];

    float* ad = agg + (size_t)d * C;
    if constexpr (C == 128) {
        const float4* ts = (const float4*)(t + (size_t)s * C);   // b128 gather
        float4 v = ts[lane];
        atomicAdd(&ad[4 * lane + 0], nrm * v.x);
        atomicAdd(&ad[4 * lane + 1], nrm * v.y);
        atomicAdd(&ad[4 * lane + 2], nrm * v.z);
        atomicAdd(&ad[4 * lane + 3], nrm * v.w);
    } else {
        const float2* ts = (const float2*)(t + (size_t)s * C);   // b64 gather
        float2 v = ts[lane];
        atomicAdd(&ad[2 * lane + 0], nrm * v.x);
        atomicAdd(&ad[2 * lane + 1], nrm * v.y);
    }
}

// ---------------- launch ----------------

static inline int cdiv(long long a, long long b) { return (int)((a + b - 1) / b); }

extern "C" void kernel_launch(void* const* d_in, const int* in_sizes, int n_in,
                              void* d_out, int out_size, void* d_ws, size_t ws_size,
                              hipStream_t stream) {
    const float* x  = (const float*)d_in[0];
    const float* W1 = (const float*)d_in[1];
    const float* b1 = (const float*)d_in[2];
    const float* W2 = (const float*)d_in[3];
    const float* b2 = (const float*)d_in[4];
    const float* W3 = (const float*)d_in[5];
    const float* b3 = (const float*)d_in[6];
    const int*   ei = (const int*)d_in[7];

    const int IN_C = 64, HID_C = 128, OUT_C = 64;
    const int n  = in_sizes[0] / IN_C;   // 65536 nodes
    const int ne = in_sizes[7] / 2;      // 1048576 edges
    const int* src = ei;
    const int* dst = ei + ne;

    // workspace: T | G (N x 128 fp32 each) | dinv (N fp32)
    float* T    = (float*)d_ws;
    float* G    = T + (size_t)n * HID_C;
    float* dinv = G + (size_t)n * HID_C;
    float* out  = (float*)d_out;

    const int TB = 256;
    const long long strip_waves = n / 16;          // one wave per 16-row strip
    const int gemm_blocks = cdiv(strip_waves * 32, TB);

    // --- degrees -> dinv ---
    deg_init_kernel<<<cdiv(n, TB), TB, 0, stream>>>(dinv, n);
    deg_count_kernel<<<cdiv(ne, TB), TB, 0, stream>>>(dst, dinv, ne);
    deg_to_dinv_kernel<<<cdiv(n, TB), TB, 0, stream>>>(dinv, n);

    // --- layer 1: T = x @ W1 ; G = agg(T) + b1 ---
    gemm_wmma_strip<false, 8><<<gemm_blocks, TB, 0, stream>>>(x, W1, T, n, IN_C);
    self_init_kernel<<<cdiv((long long)n * HID_C, TB), TB, 0, stream>>>(T, dinv, b1, G, n, HID_C);
    edge_agg_kernel<HID_C><<<cdiv((long long)ne * 32, TB), TB, 0, stream>>>(T, dinv, src, dst, G, ne);

    // --- layer 2: T = relu(G) @ W2 ; G = agg(T) + b2 ---
    gemm_wmma_strip<true, 8><<<gemm_blocks, TB, 0, stream>>>(G, W2, T, n, HID_C);
    self_init_kernel<<<cdiv((long long)n * HID_C, TB), TB, 0, stream>>>(T, dinv, b2, G, n, HID_C);
    edge_agg_kernel<HID_C><<<cdiv((long long)ne * 32, TB), TB, 0, stream>>>(T, dinv, src, dst, G, ne);

    // --- layer 3: T = relu(G) @ W3 ; d_out = agg(T) + b3 ---
    gemm_wmma_strip<true, 4><<<gemm_blocks, TB, 0, stream>>>(G, W3, T, n, HID_C);
    self_init_kernel<<<cdiv((long long)n * OUT_C, TB), TB, 0, stream>>>(T, dinv, b3, out, n, OUT_C);
    edge_agg_kernel<OUT_C><<<cdiv((long long)ne * 32, TB), TB, 0, stream>>>(T, dinv, src, dst, out, ne);
}